// DivergenceFreeMLP_89988154786346
// MI455X (gfx1250) — compile-verified
//
#include <hip/hip_runtime.h>

typedef float v2f __attribute__((ext_vector_type(2)));
typedef float v4f __attribute__((ext_vector_type(4)));
typedef float v8f __attribute__((ext_vector_type(8)));
typedef int   v4i __attribute__((ext_vector_type(4)));

#define D_   2048
#define H1_  128
#define H2_  64
#define OUT_ 512
#define B_   8192

// D = A(16x4) * B(4x16) + C, fp32 WMMA (wave32).
__device__ __forceinline__ v8f wmma_f32(v2f a, v2f b, v8f c) {
  return __builtin_amdgcn_wmma_f32_16x16x4_f32(false, a, false, b, (short)0, c,
                                               false, false);
}

// ---- gfx1250 async global->LDS copy (ASYNCcnt) with portable fallback ------
#if __has_builtin(__builtin_amdgcn_global_load_async_to_lds_b128) && \
    __has_builtin(__builtin_amdgcn_s_wait_asynccnt)
#define USE_ASYNC_LDS 1
__device__ __forceinline__ void cp16_to_lds(const float* g, float* l) {
  // AS-qualified pointers via integer round-trip (generic->AS(1) is an
  // identity mapping; generic LDS addr truncates to the AS(3) offset).
  auto gp = reinterpret_cast<__attribute__((address_space(1))) v4i*>(
      reinterpret_cast<uintptr_t>(g));
  auto lp = reinterpret_cast<__attribute__((address_space(3))) v4i*>(
      reinterpret_cast<uintptr_t>(l));
  __builtin_amdgcn_global_load_async_to_lds_b128(gp, lp, 0, 0);
}
#define WAIT_ASYNC(n) __builtin_amdgcn_s_wait_asynccnt(n)
#else
#define USE_ASYNC_LDS 0
__device__ __forceinline__ void cp16_to_lds(const float* g, float* l) {
  *(v4f*)l = *(const v4f*)g;  // synchronous fallback
}
#define WAIT_ASYNC(n) ((void)0)
#endif

// ---------------------------------------------------------------------------
// Prep: w1_rows[h] = sum_k W1[h,k];  w3_cols[j] = sum_d W3[d,j]
// ---------------------------------------------------------------------------
__global__ __launch_bounds__(256) void prep_sums_kernel(
    const float* __restrict__ W1, const float* __restrict__ W3,
    float* __restrict__ w1_rows, float* __restrict__ w3_cols) {
  __shared__ float red[256];
  const int tid = threadIdx.x;
  const int b = blockIdx.x;
  float s = 0.0f;
  if (b < H1_) {
    const float* r = W1 + (size_t)b * D_;
    for (int i = tid; i < D_; i += 256) s += r[i];
  } else {
    const int c = b - H1_;
    for (int i = tid; i < D_; i += 256) s += W3[(size_t)i * H2_ + c];
  }
  red[tid] = s;
  __syncthreads();
  for (int off = 128; off > 0; off >>= 1) {
    if (tid < off) red[tid] += red[tid + off];
    __syncthreads();
  }
  if (tid == 0) {
    if (b < H1_) w1_rows[b] = red[0];
    else         w3_cols[b - H1_] = red[0];
  }
}

// ---------------------------------------------------------------------------
// C1 = W1 @ Wp^T   [H1, OUT], K = D.  One wave per 16x16 tile.
// ---------------------------------------------------------------------------
__global__ __launch_bounds__(32) void gemm_C1_kernel(
    const float* __restrict__ W1, const float* __restrict__ Wp,
    float* __restrict__ C1) {
  const int lane  = threadIdx.x & 31;
  const int lrow  = lane & 15;
  const int khalf = lane >> 4;
  const int kd    = khalf * 2;
  const int n0 = (blockIdx.x & 31) * 16;  // 32 tiles over OUT
  const int m0 = (blockIdx.x >> 5) * 16;  // 8 tiles over H1
  const float* Ap = W1 + (size_t)(m0 + lrow) * D_ + kd;
  const float* Bp = Wp + (size_t)(n0 + lrow) * D_ + kd;
  v8f acc = {};
  for (int k = 0; k < D_; k += 4) {
    v2f a = *(const v2f*)(Ap + k);
    v2f b = *(const v2f*)(Bp + k);
    acc = wmma_f32(a, b, acc);
  }
#pragma unroll
  for (int v = 0; v < 8; ++v) {
    const int m = m0 + v + 8 * khalf;
    C1[(size_t)m * OUT_ + n0 + lrow] = acc[v];
  }
}

// ---------------------------------------------------------------------------
// C2 = Wp @ W3     [OUT, H2], K = D.  One wave per 16x16 tile.
// ---------------------------------------------------------------------------
__global__ __launch_bounds__(32) void gemm_C2_kernel(
    const float* __restrict__ Wp, const float* __restrict__ W3,
    float* __restrict__ C2) {
  const int lane  = threadIdx.x & 31;
  const int lrow  = lane & 15;
  const int khalf = lane >> 4;
  const int kd    = khalf * 2;
  const int n0 = (blockIdx.x & 3) * 16;   // 4 tiles over H2
  const int m0 = (blockIdx.x >> 2) * 16;  // 32 tiles over OUT
  const float* Ap = Wp + (size_t)(m0 + lrow) * D_ + kd;
  v8f acc = {};
  for (int k = 0; k < D_; k += 4) {
    v2f a = *(const v2f*)(Ap + k);
    v2f b;
    b.x = W3[(size_t)(k + kd) * H2_ + n0 + lrow];
    b.y = W3[(size_t)(k + kd + 1) * H2_ + n0 + lrow];
    acc = wmma_f32(a, b, acc);
  }
#pragma unroll
  for (int v = 0; v < 8; ++v) {
    const int m = m0 + v + 8 * khalf;
    C2[(size_t)m * H2_ + n0 + lrow] = acc[v];
  }
}

// ---------------------------------------------------------------------------
// Fused main: 16 batch rows per workgroup (8 wave32).
// out = s2 @ C2^T + (-s1) @ C1 + bp
// x tile is streamed into LDS via async global->LDS b128 copies (double
// buffered k-blocks of 256 floats/row), consumed by WMMA from LDS.
// ---------------------------------------------------------------------------
#define SA1 132   // padded stride for [16][128] LDS tiles (banks=64)
#define SA2 68    // padded stride for [16][64]  LDS tiles
#define XKB 256   // x k-block (floats per row per block)
#define NKB (D_ / XKB)
#define XST 268   // padded LDS row stride for x blocks (268*4 % 16 == 0)

__global__ __launch_bounds__(256) void fused_kernel(
    const float* __restrict__ x,  const float* __restrict__ W1,
    const float* __restrict__ b1, const float* __restrict__ W2,
    const float* __restrict__ b2, const float* __restrict__ bp,
    const float* __restrict__ w1_rows, const float* __restrict__ w3_cols,
    const float* __restrict__ C1, const float* __restrict__ C2,
    float* __restrict__ out) {
  __shared__ float xbuf[2][16 * XST];  // double-buffered x k-blocks
  __shared__ float a1s[16 * SA1];      // relu(h1)
  __shared__ float t1s[16 * SA1];      // m1 * w1_rows
  __shared__ float m1s[16 * SA1];      // m1
  __shared__ float s1s[16 * SA1];      // -s1
  __shared__ float s2s[16 * SA2];      // s2
  __shared__ float t2s[16 * SA2];      // m2 * w3_cols

  const int tid   = threadIdx.x;
  const int wave  = tid >> 5;
  const int lane  = tid & 31;
  const int lrow  = lane & 15;
  const int khalf = lane >> 4;
  const int kd    = khalf * 2;
  const int b0    = blockIdx.x * 16;

  // Per-block cooperative async copy: 16 rows x 256 floats = 1024 x 16B
  // chunks; each of the 256 threads issues 4 async b128 ops.
  auto issue_xblock = [&](int kb, float* dst) {
#pragma unroll
    for (int i = 0; i < 4; ++i) {
      const int q = tid + i * 256;
      const int r = q >> 6;        // row 0..15
      const int c = (q & 63) * 4;  // float offset within row (16B chunks)
      cp16_to_lds(x + (size_t)(b0 + r) * D_ + kb * XKB + c,
                  dst + r * XST + c);
    }
  };

  // ---- Phase 1: h1 = x @ W1^T + b1 ; one 16-col tile per wave (K = 2048)
  {
    const int n0 = wave * 16;
    const float* Bp = W1 + (size_t)(n0 + lrow) * D_ + kd;
    v8f acc = {};
    issue_xblock(0, xbuf[0]);
    for (int kb = 0; kb < NKB; ++kb) {
      if (kb + 1 < NKB) {
        issue_xblock(kb + 1, xbuf[(kb + 1) & 1]);
        WAIT_ASYNC(4);  // current block retired; next (4 ops) in flight
      } else {
        WAIT_ASYNC(0);  // final block fully retired
      }
      __syncthreads();
      const float* Ar = &xbuf[kb & 1][lrow * XST + kd];
      const float* Bk = Bp + kb * XKB;
      for (int k = 0; k < XKB; k += 4) {
        v2f a = *(const v2f*)(Ar + k);
        v2f b = *(const v2f*)(Bk + k);
        acc = wmma_f32(a, b, acc);
      }
      __syncthreads();  // buffer free before it is overwritten next iter
    }
    const float bias = b1[n0 + lrow];
    const float w1r  = w1_rows[n0 + lrow];
    const int   col  = n0 + lrow;
#pragma unroll
    for (int v = 0; v < 8; ++v) {
      const int m = v + 8 * khalf;
      const float h = acc[v] + bias;
      const bool pos = h > 0.0f;
      a1s[m * SA1 + col] = pos ? h : 0.0f;
      t1s[m * SA1 + col] = pos ? w1r : 0.0f;
      m1s[m * SA1 + col] = pos ? 1.0f : 0.0f;
    }
  }
  __syncthreads();

  // ---- Phase 2 (waves 0..3): h2 = relu(h1)@W2^T + b2 ; r2 = t1@W2^T (K=128)
  if (wave < 4) {
    const int n0 = wave * 16;
    const float* Bp = W2 + (size_t)(n0 + lrow) * H1_ + kd;
    v8f accH = {};
    v8f accR = {};
    for (int k = 0; k < H1_; k += 4) {
      v2f b  = *(const v2f*)(Bp + k);
      v2f aH = *(const v2f*)(&a1s[lrow * SA1 + k + kd]);
      v2f aR = *(const v2f*)(&t1s[lrow * SA1 + k + kd]);
      accH = wmma_f32(aH, b, accH);
      accR = wmma_f32(aR, b, accR);
    }
    const float bias = b2[n0 + lrow];
    const float w3c  = w3_cols[n0 + lrow];
    const int   col  = n0 + lrow;
#pragma unroll
    for (int v = 0; v < 8; ++v) {
      const int m = v + 8 * khalf;
      const float h2 = accH[v] + bias;
      const bool pos = h2 > 0.0f;
      s2s[m * SA2 + col] = pos ? accR[v] : 0.0f;
      t2s[m * SA2 + col] = pos ? w3c : 0.0f;
    }
  }
  __syncthreads();

  // ---- Phase 3: s1n = -(t2 @ W2) * m1 ; one 16-col tile per wave (K = 64)
  {
    const int n0 = wave * 16;
    v8f acc = {};
    for (int k = 0; k < H2_; k += 4) {
      v2f a = *(const v2f*)(&t2s[lrow * SA2 + k + kd]);
      v2f b;
      b.x = W2[(size_t)(k + kd) * H1_ + n0 + lrow];
      b.y = W2[(size_t)(k + kd + 1) * H1_ + n0 + lrow];
      acc = wmma_f32(a, b, acc);
    }
    const int col = n0 + lrow;
#pragma unroll
    for (int v = 0; v < 8; ++v) {
      const int m = v + 8 * khalf;
      s1s[m * SA1 + col] = -acc[v] * m1s[m * SA1 + col];
    }
  }
  __syncthreads();

  // ---- Phase 4: out = s2@C2^T + s1n@C1 + bp ; 4 output tiles per wave
#pragma unroll
  for (int t = 0; t < 4; ++t) {
    const int n0 = (wave * 4 + t) * 16;
    v8f acc = {};
    // s2 @ C2^T (K = 64), C2 row-major along K
    const float* Bp2 = C2 + (size_t)(n0 + lrow) * H2_ + kd;
    for (int k = 0; k < H2_; k += 4) {
      v2f a = *(const v2f*)(&s2s[lrow * SA2 + k + kd]);
      v2f b = *(const v2f*)(Bp2 + k);
      acc = wmma_f32(a, b, acc);
    }
    // s1n @ C1 (K = 128), C1 strided along K
    for (int k = 0; k < H1_; k += 4) {
      v2f a = *(const v2f*)(&s1s[lrow * SA1 + k + kd]);
      v2f b;
      b.x = C1[(size_t)(k + kd) * OUT_ + n0 + lrow];
      b.y = C1[(size_t)(k + kd + 1) * OUT_ + n0 + lrow];
      acc = wmma_f32(a, b, acc);
    }
    const float bias = bp[n0 + lrow];
#pragma unroll
    for (int v = 0; v < 8; ++v) {
      const int m = v + 8 * khalf;
      out[(size_t)(b0 + m) * OUT_ + n0 + lrow] = acc[v] + bias;
    }
  }
}

// ---------------------------------------------------------------------------
extern "C" void kernel_launch(void* const* d_in, const int* in_sizes, int n_in,
                              void* d_out, int out_size, void* d_ws,
                              size_t ws_size, hipStream_t stream) {
  const float* x  = (const float*)d_in[0];
  const float* W1 = (const float*)d_in[1];
  const float* b1 = (const float*)d_in[2];
  const float* W2 = (const float*)d_in[3];
  const float* b2 = (const float*)d_in[4];
  const float* W3 = (const float*)d_in[5];
  // d_in[6] = b3 : unused by the reference output
  const float* Wp = (const float*)d_in[7];
  const float* bp = (const float*)d_in[8];
  float* out = (float*)d_out;

  float* ws       = (float*)d_ws;
  float* w1_rows  = ws;                    // 128
  float* w3_cols  = ws + 128;              // 64
  float* C1       = ws + 192;              // 128*512
  float* C2       = C1 + H1_ * OUT_;       // 512*64

  prep_sums_kernel<<<dim3(H1_ + H2_), dim3(256), 0, stream>>>(W1, W3, w1_rows,
                                                              w3_cols);
  gemm_C1_kernel<<<dim3(256), dim3(32), 0, stream>>>(W1, Wp, C1);
  gemm_C2_kernel<<<dim3(128), dim3(32), 0, stream>>>(Wp, W3, C2);
  fused_kernel<<<dim3(B_ / 16), dim3(256), 0, stream>>>(
      x, W1, b1, W2, b2, bp, w1_rows, w3_cols, C1, C2, out);
}